// SoftClustering_39195871543495
// MI455X (gfx1250) — compile-verified
//
#include <hip/hip_runtime.h>
#include <hip/hip_bf16.h>

// CDNA5 / gfx1250 fused soft-clustering kernel.
// FP32 WMMA (v_wmma_f32_16x16x4_f32) for both the Gram matrix and the
// weights@codebook GEMM; sim matrix is never materialized.

#define N_CODES 8192
#define FEAT    64
#define NTILES  (N_CODES / 16)   // 512 column tiles of 16
#define WAVES_PER_BLOCK 4

typedef float v2f __attribute__((ext_vector_type(2)));
typedef float v8f __attribute__((ext_vector_type(8)));

#define LOG2E 1.44269504088896340736f

// fast, safe-for-our-range sim(d2) = exp(-sqrt(d2)); d2 >= 1e-12 (no denorms),
// exp2 argument >= ~-40 (no underflow issues vs fp32 tolerance).
__device__ __forceinline__ float sim_of_d2(float d2) {
    float dist = __builtin_amdgcn_sqrtf(d2);          // v_sqrt_f32
    return __builtin_amdgcn_exp2f(-LOG2E * dist);     // v_exp_f32
}

// ---------------------------------------------------------------------------
// Pre-kernel: per-row squared L2 norms  sq[i] = sum_k C[i][k]^2
// ---------------------------------------------------------------------------
__global__ __launch_bounds__(256) void sq_kernel(const float* __restrict__ cb,
                                                 float* __restrict__ sq) {
    int i = blockIdx.x * blockDim.x + threadIdx.x;
    if (i < N_CODES) {
        const float4* row = (const float4*)(cb + (size_t)i * FEAT);
        float s = 0.0f;
#pragma unroll
        for (int k = 0; k < FEAT / 4; ++k) {
            float4 v = row[k];
            s += v.x * v.x + v.y * v.y + v.z * v.z + v.w * v.w;
        }
        sq[i] = s;
    }
}

// ---------------------------------------------------------------------------
// Main fused kernel: one wave owns one 16-row tile.
//  Sweep 1 over all 512 column tiles: WMMA gram -> sim -> rowsum (registers).
//  Sweep 2: recompute sim, scale by 1/rowsum, LDS-transpose the 16x16 weight
//  tile into WMMA A-layout, accumulate weights @ codebook with WMMA.
// ---------------------------------------------------------------------------
__global__ __launch_bounds__(32 * WAVES_PER_BLOCK) void
soft_cluster_kernel(const float* __restrict__ cb,
                    const float* __restrict__ delta_p,
                    const float* __restrict__ sq,
                    float* __restrict__ out) {
    // Padded wave-private 16x16 transpose scratch (stride 17 => conflict-free
    // column reads since gcd(17,64)==1).
    __shared__ float ldsw[WAVES_PER_BLOCK][16][17];

    const int lane = threadIdx.x & 31;
    const int wid  = threadIdx.x >> 5;
    const int tile = blockIdx.x * WAVES_PER_BLOCK + wid;   // 0..511
    const int ibase = tile * 16;
    const int half  = lane >> 4;    // 0: lanes 0-15, 1: lanes 16-31
    const int l16   = lane & 15;

    // --- Load the 16x64 row tile of the codebook into WMMA A-layout regs.
    // A (16x4 fp32) per k-step kk: lane holds M = l16, K = 4*kk + 2*half + {0,1}.
    v2f A[16];
    {
        const float* arow = cb + (size_t)(ibase + l16) * FEAT + 2 * half;
#pragma unroll
        for (int kk = 0; kk < 16; ++kk)
            A[kk] = *(const v2f*)(arow + 4 * kk);
    }

    // sq of my rows: C/D-layout VGPR r holds row M = r + 8*half.
    float sqi[8];
#pragma unroll
    for (int r = 0; r < 8; ++r) sqi[r] = sq[ibase + r + 8 * half];

    // ======================= Sweep 1: row sums ============================
    float prs[8];
#pragma unroll
    for (int r = 0; r < 8; ++r) prs[r] = 0.0f;

    for (int jt = 0; jt < NTILES; ++jt) {
        const int jbase = jt * 16;
        const float* brow = cb + (size_t)(jbase + l16) * FEAT + 2 * half;
        // Prefetch next tile's operand row (one cacheline per lane).
        if (jt + 1 < NTILES)
            __builtin_prefetch(brow + FEAT * 16, 0, 3);

        // Two interleaved accumulator chains for XDL pipelining.
        v8f c0 = {}, c1 = {};
#pragma unroll
        for (int kk = 0; kk < 16; kk += 2) {
            v2f b0 = *(const v2f*)(brow + 4 * kk);
            v2f b1 = *(const v2f*)(brow + 4 * kk + 4);
            c0 = __builtin_amdgcn_wmma_f32_16x16x4_f32(
                    false, A[kk], false, b0, (short)0, c0, false, false);
            c1 = __builtin_amdgcn_wmma_f32_16x16x4_f32(
                    false, A[kk + 1], false, b1, (short)0, c1, false, false);
        }
        const v8f c = c0 + c1;
        const float sqj = sq[jbase + l16];
#pragma unroll
        for (int r = 0; r < 8; ++r) {
            float d2 = fmaxf(sqi[r] + sqj - 2.0f * c[r], 1e-12f);
            prs[r] += sim_of_d2(d2);
        }
    }

    // Reduce each row-sum across its 16 lanes (masks 1,2,4,8 stay in-half).
#pragma unroll
    for (int r = 0; r < 8; ++r) {
        float v = prs[r];
#pragma unroll
        for (int m = 1; m < 16; m <<= 1) v += __shfl_xor(v, m, 32);
        prs[r] = 1.0f / v;   // inverse row sum, uniform within each half
    }

    // ================= Sweep 2: weighted sum accumulation =================
    v8f acc[4] = {{}, {}, {}, {}};   // 16 rows x 64 cols in 4 C/D tiles

    for (int jt = 0; jt < NTILES; ++jt) {
        const int jbase = jt * 16;
        const float* brow = cb + (size_t)(jbase + l16) * FEAT + 2 * half;
        if (jt + 1 < NTILES)
            __builtin_prefetch(brow + FEAT * 16, 0, 3);

        v8f c0 = {}, c1 = {};
#pragma unroll
        for (int kk = 0; kk < 16; kk += 2) {
            v2f b0 = *(const v2f*)(brow + 4 * kk);
            v2f b1 = *(const v2f*)(brow + 4 * kk + 4);
            c0 = __builtin_amdgcn_wmma_f32_16x16x4_f32(
                    false, A[kk], false, b0, (short)0, c0, false, false);
            c1 = __builtin_amdgcn_wmma_f32_16x16x4_f32(
                    false, A[kk + 1], false, b1, (short)0, c1, false, false);
        }
        const v8f c = c0 + c1;
        const float sqj = sq[jbase + l16];

        // Normalized weights -> LDS in [row][col] order (C/D layout write).
#pragma unroll
        for (int r = 0; r < 8; ++r) {
            float d2 = fmaxf(sqi[r] + sqj - 2.0f * c[r], 1e-12f);
            float w  = sim_of_d2(d2) * prs[r];
            ldsw[wid][r + 8 * half][l16] = w;
        }
        // DS ops are in-order within a wave; this is a compiler barrier +
        // drain so the cross-lane transpose reads see the writes.
        asm volatile("s_wait_dscnt 0x0" ::: "memory");

        // Read weight tile back in WMMA A-layout: M = l16, K = 4kk+2half+{0,1}.
        v2f W[4];
#pragma unroll
        for (int kk = 0; kk < 4; ++kk) {
            W[kk].x = ldsw[wid][l16][4 * kk + 2 * half + 0];
            W[kk].y = ldsw[wid][l16][4 * kk + 2 * half + 1];
        }
        asm volatile("" ::: "memory");  // keep reads before next-iter writes

        // acc[nt] += W(16x16) @ C[jbase:jbase+16, nt*16:nt*16+16]
        // 4 independent accumulator chains (one per N-tile) => good XDL ILP.
#pragma unroll
        for (int nt = 0; nt < 4; ++nt) {
#pragma unroll
            for (int kk = 0; kk < 4; ++kk) {
                const int k0 = jbase + 4 * kk + 2 * half;
                v2f b;
                b.x = cb[(size_t)(k0 + 0) * FEAT + nt * 16 + l16];
                b.y = cb[(size_t)(k0 + 1) * FEAT + nt * 16 + l16];
                acc[nt] = __builtin_amdgcn_wmma_f32_16x16x4_f32(
                              false, W[kk], false, b, (short)0, acc[nt],
                              false, false);
            }
        }
    }

    // ============================ Epilogue ================================
    const float delta = *delta_p;
    const float omd   = 1.0f - delta;
#pragma unroll
    for (int nt = 0; nt < 4; ++nt) {
#pragma unroll
        for (int r = 0; r < 8; ++r) {
            const int row = ibase + r + 8 * half;
            const int col = nt * 16 + l16;
            const size_t idx = (size_t)row * FEAT + col;
            out[idx] = omd * cb[idx] + delta * acc[nt][r];
        }
    }
}

// ---------------------------------------------------------------------------
extern "C" void kernel_launch(void* const* d_in, const int* in_sizes, int n_in,
                              void* d_out, int out_size, void* d_ws, size_t ws_size,
                              hipStream_t stream) {
    const float* cb    = (const float*)d_in[0];
    const float* delta = (const float*)d_in[1];
    float* out = (float*)d_out;
    float* sq  = (float*)d_ws;          // 8192 floats = 32 KB scratch

    hipLaunchKernelGGL(sq_kernel, dim3(N_CODES / 256), dim3(256), 0, stream,
                       cb, sq);

    // 512 row tiles, 1 wave each: 128 blocks x 4 waves.
    hipLaunchKernelGGL(soft_cluster_kernel,
                       dim3(NTILES / WAVES_PER_BLOCK),
                       dim3(32 * WAVES_PER_BLOCK), 0, stream,
                       cb, delta, sq, out);
}